// DifferentiableKalmanCell_22368189677917
// MI455X (gfx1250) — compile-verified
//
#include <hip/hip_runtime.h>
#include <hip/hip_bf16.h>

// Sizes from the reference
#define BN        4096
#define SDIM      64
#define ODIM      32
#define NCO       11      // BASE_DEGREE + 3 coefficients stored
#define TSTEPS    96
#define EPSV      1e-7f

#define P64       68      // LDS pitch for 64-wide matrices (16B-aligned, conflict-free)
#define P32       36      // LDS pitch for 32-wide matrices

typedef float v2f __attribute__((ext_vector_type(2)));
typedef float v8f __attribute__((ext_vector_type(8)));

// One 16x16 f32 tile: acc += A(16xKD) * B(KDx16), operands in LDS.
// A row-major with pitch lda. If BT, B is stored transposed (Bm[n][k], pitch ldb).
// WMMA f32 16x16x4 operand layout (ISA 7.12.2):
//   A: lanes0-15 m=lane {k,k+1}; lanes16-31 m=lane-16 {k+2,k+3}
//   B: V0 = rows k (lanes0-15) / k+2 (lanes16-31); V1 = rows k+1 / k+3
template<bool BT, int KD>
__device__ __forceinline__ v8f mm16(const float* __restrict__ A, int lda,
                                    const float* __restrict__ Bm, int ldb,
                                    int tm, int tn, v8f acc, int l16, int hh) {
#pragma unroll
  for (int k0 = 0; k0 < KD; k0 += 4) {
    const int ks = k0 + hh * 2;
    v2f a, b;
    const float* ap = A + (tm + l16) * lda + ks;
    a.x = ap[0];
    a.y = ap[1];
    if (!BT) {
      b.x = Bm[(ks + 0) * ldb + tn + l16];
      b.y = Bm[(ks + 1) * ldb + tn + l16];
    } else {
      const float* bp = Bm + (tn + l16) * ldb + ks;
      b.x = bp[0];
      b.y = bp[1];
    }
    acc = __builtin_amdgcn_wmma_f32_16x16x4_f32(false, a, false, b, (short)0,
                                                acc, false, false);
  }
  return acc;
}

// C/D layout: VGPR r -> row tm+r (lanes0-15) / tm+r+8 (lanes16-31), col tn+lane16
__device__ __forceinline__ void stile(float* __restrict__ C, int ldc,
                                      int tm, int tn, v8f acc, int l16, int hh) {
#pragma unroll
  for (int r = 0; r < 8; ++r)
    C[(tm + r + hh * 8) * ldc + tn + l16] = acc[r];
}

__global__ __launch_bounds__(256)
void DifferentiableKalmanCell_kernel(const float* __restrict__ obs,
                                     const float* __restrict__ prev_state,
                                     const float* __restrict__ prev_cov,
                                     const float* __restrict__ coeffs,
                                     const float* __restrict__ obs_matrix,
                                     const float* __restrict__ logQ,
                                     const float* __restrict__ logR,
                                     const int* __restrict__ tptr,
                                     float* __restrict__ out_state,
                                     float* __restrict__ out_cov,
                                     float* __restrict__ out_diag) {
  __shared__ __align__(16) float Pm[SDIM * P64];   // prev_cov, later U
  __shared__ __align__(16) float Fm[SDIM * P64];   // Fj
  __shared__ __align__(16) float Tm[SDIM * P64];   // T = Fj*P, later I-KH
  __shared__ __align__(16) float Dm[SDIM * P64];   // cov_pred, later cov_update
  __shared__ __align__(16) float Hs[ODIM * P64];   // H (32x64)
  __shared__ __align__(16) float PHt[SDIM * P32];  // cov_pred * H^T
  __shared__ __align__(16) float Ks[SDIM * P32];   // Kalman gain
  __shared__ __align__(16) float Krs[SDIM * P32];  // K * diag(r)
  __shared__ __align__(16) float Ss[ODIM * P32];   // S, then Cholesky L in place
  __shared__ __align__(16) float Li[ODIM * P32];   // L^-1
  __shared__ __align__(16) float Si[ODIM * P32];   // S^-1
  __shared__ float chebS[SDIM * 12];
  __shared__ float dchS[SDIM * 12];
  __shared__ float spS[SDIM];
  __shared__ float qS[SDIM];
  __shared__ float rS[ODIM];
  __shared__ float resS[ODIM];

  const int tid = threadIdx.x;
  const int bb  = blockIdx.x;
  const int lane = tid & 31;
  const int wv   = tid >> 5;
  const int l16  = lane & 15;
  const int hh   = lane >> 4;

  // ---- degree(t), computed on-device (t is a device scalar) ----
  const int tval = tptr[0];
  const float pfrac = (float)tval / (float)(TSTEPS - 1);
  const float scaled = 0.5f * (1.0f + sinf(6.28318530717958647692f * pfrac));
  int deg = 8 + (int)(2.0f * scaled);
  if (deg > 10) deg = 10;
  if (deg < 8)  deg = 8;

  // ---- stage 0: global -> LDS ----
  {
    const float* pc = prev_cov + (size_t)bb * (SDIM * SDIM);
    for (int idx = tid; idx < (SDIM * SDIM / 4); idx += 256) {
      const int row = idx >> 4, c4 = (idx & 15) << 2;
      *(float4*)(Pm + row * P64 + c4) = *(const float4*)(pc + row * SDIM + c4);
    }
    for (int idx = tid; idx < (ODIM * SDIM / 4); idx += 256) {
      const int row = idx >> 4, c4 = (idx & 15) << 2;
      *(float4*)(Hs + row * P64 + c4) = *(const float4*)(obs_matrix + row * SDIM + c4);
    }
    if (tid < SDIM) qS[tid] = expf(logQ[tid]);
    if (tid < ODIM) rS[tid] = expf(logR[tid]);

    if (tid < SDIM) {
      const float x  = prev_state[(size_t)bb * SDIM + tid];
      const float tr = tanhf(x);
      float tv = tr;
      if (tv < -1.0f + EPSV) tv = -1.0f + EPSV;
      if (tv >  1.0f - EPSV) tv =  1.0f - EPSV;
      const float th   = acosf(tv);
      const float mask = (tr > -1.0f + EPSV && tr < 1.0f - EPSV) ? 1.0f : 0.0f;
      const float dth  = -(1.0f - tr * tr) * mask / sqrtf(1.0f - tv * tv);
      for (int d = 0; d <= deg; ++d) {
        const float ang = (float)d * th;
        chebS[tid * 12 + d] = cosf(ang);
        dchS[tid * 12 + d]  = -(float)d * sinf(ang) * dth;
      }
    }
  }
  __syncthreads();

  // ---- stage 1: Fj[o][i] and state_pred ----
  for (int idx = tid; idx < SDIM * SDIM; idx += 256) {
    const int o = idx & 63, i = idx >> 6;
    const float* cp = coeffs + (i * SDIM + o) * NCO;
    float s = 0.0f;
    for (int d = 0; d <= deg; ++d) s += dchS[i * 12 + d] * cp[d];
    Fm[o * P64 + i] = s;
  }
  if (tid < SDIM) {
    const int o = tid;
    float s = 0.0f;
    for (int i = 0; i < SDIM; ++i) {
      const float* cp = coeffs + (i * SDIM + o) * NCO;
      for (int d = 0; d <= deg; ++d) s += chebS[i * 12 + d] * cp[d];
    }
    spS[o] = s;
  }
  __syncthreads();

  // ---- stage 2: residual (VALU) and T = Fj * P (WMMA) ----
  if (tid < ODIM) {
    float s = 0.0f;
    for (int j = 0; j < SDIM; ++j) s += Hs[tid * P64 + j] * spS[j];
    resS[tid] = obs[(size_t)bb * ODIM + tid] - s;
  }
  for (int t = wv; t < 16; t += 8) {
    const int tm = (t >> 2) * 16, tn = (t & 3) * 16;
    v8f acc = {};
    acc = mm16<false, 64>(Fm, P64, Pm, P64, tm, tn, acc, l16, hh);
    stile(Tm, P64, tm, tn, acc, l16, hh);
  }
  __syncthreads();

  // ---- stage 3: cov_pred = T * Fj^T + diag(q) ----
  for (int t = wv; t < 16; t += 8) {
    const int tm = (t >> 2) * 16, tn = (t & 3) * 16;
    v8f acc = {};
    acc = mm16<true, 64>(Tm, P64, Fm, P64, tm, tn, acc, l16, hh);
#pragma unroll
    for (int r = 0; r < 8; ++r) {
      const int row = tm + r + hh * 8, col = tn + l16;
      float v = acc[r];
      if (row == col) v += qS[row];
      Dm[row * P64 + col] = v;
    }
  }
  __syncthreads();

  // ---- stage 4: PHt = cov_pred * H^T (64x32) ----
  {
    const int tm = (wv >> 1) * 16, tn = (wv & 1) * 16;
    v8f acc = {};
    acc = mm16<true, 64>(Dm, P64, Hs, P64, tm, tn, acc, l16, hh);
    stile(PHt, P32, tm, tn, acc, l16, hh);
  }
  __syncthreads();

  // ---- stage 5: S = H * PHt + diag(r) (32x32) ----
  if (wv < 4) {
    const int tm = (wv >> 1) * 16, tn = (wv & 1) * 16;
    v8f acc = {};
    acc = mm16<false, 64>(Hs, P64, PHt, P32, tm, tn, acc, l16, hh);
#pragma unroll
    for (int r = 0; r < 8; ++r) {
      const int row = tm + r + hh * 8, col = tn + l16;
      float v = acc[r];
      if (row == col) v += rS[row];
      Ss[row * P32 + col] = v;
    }
  }
  __syncthreads();

  // ---- stage 6: S^-1 via Cholesky (S is SPD) ----
  for (int k = 0; k < ODIM; ++k) {
    if (tid == 0) Ss[k * P32 + k] = sqrtf(Ss[k * P32 + k]);
    __syncthreads();
    if (tid > k && tid < ODIM) Ss[tid * P32 + k] /= Ss[k * P32 + k];
    __syncthreads();
    for (int idx = tid; idx < ODIM * ODIM; idx += 256) {
      const int i = idx >> 5, j = idx & 31;
      if (i > k && j > k && j <= i)
        Ss[i * P32 + j] -= Ss[i * P32 + k] * Ss[j * P32 + k];
    }
    __syncthreads();
  }
  if (tid < ODIM) {            // columns of L^-1, independent per thread
    const int c = tid;
    for (int i = 0; i < ODIM; ++i) {
      float s;
      if (i < c) s = 0.0f;
      else {
        s = (i == c) ? 1.0f : 0.0f;
        for (int j = c; j < i; ++j) s -= Ss[i * P32 + j] * Li[j * P32 + c];
        s /= Ss[i * P32 + i];
      }
      Li[i * P32 + c] = s;
    }
  }
  __syncthreads();
  for (int idx = tid; idx < ODIM * ODIM; idx += 256) {  // S^-1 = L^-T L^-1
    const int a = idx >> 5, c = idx & 31;
    float s = 0.0f;
    for (int i = 0; i < ODIM; ++i) s += Li[i * P32 + a] * Li[i * P32 + c];
    Si[a * P32 + c] = s;
  }
  __syncthreads();

  // ---- stage 7: K = PHt * S^-1 (64x32) ----
  {
    const int tm = (wv >> 1) * 16, tn = (wv & 1) * 16;
    v8f acc = {};
    acc = mm16<false, 32>(PHt, P32, Si, P32, tm, tn, acc, l16, hh);
    stile(Ks, P32, tm, tn, acc, l16, hh);
  }
  __syncthreads();

  // ---- stage 8: Kr = K*diag(r); state_update out ----
  for (int idx = tid; idx < SDIM * ODIM; idx += 256) {
    const int i = idx >> 5, o = idx & 31;
    Krs[i * P32 + o] = Ks[i * P32 + o] * rS[o];
  }
  if (tid < SDIM) {
    float s = spS[tid];
    for (int o = 0; o < ODIM; ++o) s += Ks[tid * P32 + o] * resS[o];
    out_state[(size_t)bb * SDIM + tid] = s;
  }
  __syncthreads();

  // ---- stage 9: IKH = I - K*H ----
  for (int t = wv; t < 16; t += 8) {
    const int tm = (t >> 2) * 16, tn = (t & 3) * 16;
    v8f acc = {};
    acc = mm16<false, 32>(Ks, P32, Hs, P64, tm, tn, acc, l16, hh);
#pragma unroll
    for (int r = 0; r < 8; ++r) {
      const int row = tm + r + hh * 8, col = tn + l16;
      Tm[row * P64 + col] = ((row == col) ? 1.0f : 0.0f) - acc[r];
    }
  }
  __syncthreads();

  // ---- stage 10: U = IKH * cov_pred ----
  for (int t = wv; t < 16; t += 8) {
    const int tm = (t >> 2) * 16, tn = (t & 3) * 16;
    v8f acc = {};
    acc = mm16<false, 64>(Tm, P64, Dm, P64, tm, tn, acc, l16, hh);
    stile(Pm, P64, tm, tn, acc, l16, hh);
  }
  __syncthreads();

  // ---- stage 11: cov_update = U * IKH^T + Kr * K^T ----
  for (int t = wv; t < 16; t += 8) {
    const int tm = (t >> 2) * 16, tn = (t & 3) * 16;
    v8f acc = {};
    acc = mm16<true, 64>(Pm, P64, Tm, P64, tm, tn, acc, l16, hh);
    acc = mm16<true, 32>(Krs, P32, Ks, P32, tm, tn, acc, l16, hh);
    stile(Dm, P64, tm, tn, acc, l16, hh);
  }
  __syncthreads();

  // ---- stage 12: write cov_update and cov_diag ----
  {
    float* oc = out_cov + (size_t)bb * (SDIM * SDIM);
    for (int idx = tid; idx < (SDIM * SDIM / 4); idx += 256) {
      const int row = idx >> 4, c4 = (idx & 15) << 2;
      *(float4*)(oc + row * SDIM + c4) = *(const float4*)(Dm + row * P64 + c4);
    }
    if (tid < SDIM) out_diag[(size_t)bb * SDIM + tid] = Dm[tid * P64 + tid];
  }
}

extern "C" void kernel_launch(void* const* d_in, const int* in_sizes, int n_in,
                              void* d_out, int out_size, void* d_ws, size_t ws_size,
                              hipStream_t stream) {
  const float* obs        = (const float*)d_in[0];
  const float* prev_state = (const float*)d_in[1];
  const float* prev_cov   = (const float*)d_in[2];
  const float* coeffs     = (const float*)d_in[3];
  const float* obs_matrix = (const float*)d_in[4];
  const float* logQ       = (const float*)d_in[5];
  const float* logR       = (const float*)d_in[6];
  const int*   tptr       = (const int*)d_in[7];

  float* out       = (float*)d_out;
  float* out_state = out;                                   // [4096,64]
  float* out_cov   = out + (size_t)BN * SDIM;               // [4096,64,64]
  float* out_diag  = out + (size_t)BN * SDIM + (size_t)BN * SDIM * SDIM; // [4096,64]

  DifferentiableKalmanCell_kernel<<<dim3(BN), dim3(256), 0, stream>>>(
      obs, prev_state, prev_cov, coeffs, obs_matrix, logQ, logR, tptr,
      out_state, out_cov, out_diag);
}